// DynamicVoxelEncoder_19318762898066
// MI455X (gfx1250) — compile-verified
//
#include <hip/hip_runtime.h>
#include <stdint.h>

// scatter_mean: out[m, c] = sum_{i : idx[i]==m} in[i, c] / max(count(m), 1)
// N = 4,000,000 rows, C = 16 channels (f32), M = 500,000 segments.
// Memory/atomic bound. Accumulator (32 MB sums + 2 MB counts) fits in MI455X's
// 192 MB L2, so atomics resolve in L2; the 256 MB input stream is read once
// with non-temporal hints so it does not thrash the resident accumulator.

#define C_CHANNELS 16

typedef float v4f __attribute__((ext_vector_type(4)));

// ---------------------------------------------------------------------------
// Kernel 1: zero the accumulator (d_out, reused as the sum buffer) and counts.
// d_out is poisoned by the harness and left dirty between replays, so this
// runs every launch. out has out_size/4 float4-quads; counts has M entries.
// ---------------------------------------------------------------------------
__global__ void dve_zero_kernel(v4f* __restrict__ out4,
                                unsigned* __restrict__ counts,
                                int n4, int M) {
  int i = blockIdx.x * blockDim.x + threadIdx.x;
  if (i < n4) {
    v4f z = {0.f, 0.f, 0.f, 0.f};
    out4[i] = z;
  }
  if (i < M) counts[i] = 0u;
}

// ---------------------------------------------------------------------------
// Kernel 2: scatter-add. Thread t handles row = t>>2, channel-quad q = t&3.
// A wave32 covers 8 consecutive rows = 512 contiguous input bytes
// (global_load_b128 per lane, fully coalesced, TH=NT since data is streamed
// exactly once). Each lane issues 4 non-returning global_atomic_add_f32 into
// the L2-resident sum buffer; quad-lane 0 bumps the segment count once/row.
// ---------------------------------------------------------------------------
__global__ void dve_scatter_kernel(const float* __restrict__ in,
                                   const int* __restrict__ idx,
                                   float* __restrict__ out,
                                   unsigned* __restrict__ counts,
                                   int nQuads) {
  int t = blockIdx.x * blockDim.x + threadIdx.x;
  if (t >= nQuads) return;
  int row = t >> 2;
  int q   = t & 3;

  // Pull the streaming input ~32 KB ahead into cache (global_prefetch_b8;
  // speculative — OOB prefetches at the tail are silently dropped).
  __builtin_prefetch(in + (size_t)row * C_CHANNELS + 512 * C_CHANNELS, 0, 1);

  // Segment id: streamed once -> non-temporal load.
  int seg = __builtin_nontemporal_load(idx + row);

  // 16 bytes of this row's channels: non-temporal b128 load.
  const v4f* inq = reinterpret_cast<const v4f*>(in) + ((size_t)row * 4 + q);
  v4f v = __builtin_nontemporal_load(inq);

  float* dst = out + (size_t)seg * C_CHANNELS + q * 4;
  // Non-returning f32 atomics -> global_atomic_add_f32 (no RET), STOREcnt.
  atomicAdd(dst + 0, v.x);
  atomicAdd(dst + 1, v.y);
  atomicAdd(dst + 2, v.z);
  atomicAdd(dst + 3, v.w);

  if (q == 0) {
    atomicAdd(counts + seg, 1u);  // one count increment per row
  }
}

// ---------------------------------------------------------------------------
// Kernel 3: finalize. Thread t handles segment m = t>>2, channel-quad q = t&3;
// the quad index into out is exactly t (m*4 + q), so loads/stores are
// coalesced b128. counts[m] is shared by the 4 lanes of a segment (L2 hit).
// ---------------------------------------------------------------------------
__global__ void dve_finalize_kernel(float* __restrict__ out,
                                    const unsigned* __restrict__ counts,
                                    int mQuads) {
  int t = blockIdx.x * blockDim.x + threadIdx.x;
  if (t >= mQuads) return;
  int m = t >> 2;
  float cnt = (float)counts[m];
  float inv = 1.0f / fmaxf(cnt, 1.0f);
  v4f* p = reinterpret_cast<v4f*>(out) + t;
  v4f v = *p;
  v.x *= inv; v.y *= inv; v.z *= inv; v.w *= inv;
  *p = v;
}

// ---------------------------------------------------------------------------
// Launch wrapper. Derives M from out_size (avoids any device read of the
// num_segments scalar) and N from in_sizes[0]. All launches on `stream`,
// no allocation / sync -> graph-capture safe and replay-deterministic
// (modulo f32 atomic ordering, inherent to scatter-add).
// ---------------------------------------------------------------------------
extern "C" void kernel_launch(void* const* d_in, const int* in_sizes, int n_in,
                              void* d_out, int out_size, void* d_ws, size_t ws_size,
                              hipStream_t stream) {
  const float* in  = (const float*)d_in[0];  // [N, 16] f32
  const int*   idx = (const int*)d_in[1];    // [N] segment ids (int, < M)
  float* out = (float*)d_out;                // [M, 16] f32
  unsigned* counts = (unsigned*)d_ws;        // [M] u32 scratch (2 MB)

  const int NC = in_sizes[0];                // N * 16
  const int N  = NC / C_CHANNELS;            // 4,000,000
  const int M  = out_size / C_CHANNELS;      // 500,000

  const int threads = 256;                   // 8 waves of 32 on CDNA5

  // 1) zero sums + counts
  int n4 = out_size / 4;                     // float4 quads in out
  int zeroWork = (n4 > M) ? n4 : M;
  dve_zero_kernel<<<(zeroWork + threads - 1) / threads, threads, 0, stream>>>(
      reinterpret_cast<v4f*>(out), counts, n4, M);

  // 2) scatter-add with L2 atomics
  int nQuads = N * 4;                        // 16M threads
  dve_scatter_kernel<<<(nQuads + threads - 1) / threads, threads, 0, stream>>>(
      in, idx, out, counts, nQuads);

  // 3) divide by clamped counts
  int mQuads = M * 4;                        // 2M threads
  dve_finalize_kernel<<<(mQuads + threads - 1) / threads, threads, 0, stream>>>(
      out, counts, mQuads);
}